// GCN_layer_31739808318040
// MI455X (gfx1250) — compile-verified
//
#include <hip/hip_runtime.h>
#include <math.h>

#define D_IN  256
#define D_OUT 128

typedef __attribute__((ext_vector_type(2))) float v2f;
typedef __attribute__((ext_vector_type(8))) float v8f;

// ---------------------------------------------------------------------------
// Float atomic-max via sign-split integer atomics (IEEE-754 order-preserving):
// non-negative floats compare correctly as int; negative floats compare in
// reverse as uint, so use atomicMin on the uint view. Init value must be -inf.
// ---------------------------------------------------------------------------
__device__ __forceinline__ void atomic_max_f32(float* addr, float val) {
    if (val >= 0.0f) {
        atomicMax((int*)addr, __float_as_int(val));
    } else {
        atomicMin((unsigned int*)addr, __float_as_uint(val));
    }
}

// ---------------------------------------------------------------------------
// Init: zero the output accumulator (float4 stores), emax = -inf, denom = 0.
// One thread handles 4 output floats.
// ---------------------------------------------------------------------------
__global__ __launch_bounds__(256) void gat_init(float4* __restrict__ out4,
                                                float* __restrict__ emax,
                                                float* __restrict__ denom,
                                                int n_nodes) {
    int i = blockIdx.x * 256 + threadIdx.x;
    int total4 = n_nodes * (D_OUT / 4);
    if (i < total4) out4[i] = make_float4(0.f, 0.f, 0.f, 0.f);
    if (i < n_nodes) {
        emax[i]  = -INFINITY;
        denom[i] = 0.0f;
    }
}

// ---------------------------------------------------------------------------
// hp = h @ W + b via V_WMMA_F32_16X16X4_F32.
// Block = 256 threads = 8 waves; block handles a 16-row node tile, wave w
// handles columns [16w, 16w+16). 64 k-steps of 4, unrolled by 8.
//
// A (16x4 f32): lanes 0-15 hold K={0,1} in v0/v1, lanes 16-31 hold K={2,3}
//   -> one b64 load per lane per step.
// B (4x16): v0 = rows K {0|2}, v1 = rows K {1|3} for lane halves.
// C/D: VGPR i -> row (i + 8*half), col = lane&15.
// FULL_TILE=true when n_nodes % 16 == 0: no clamping, branch-free epilogue.
// ---------------------------------------------------------------------------
template<bool FULL_TILE>
__global__ __launch_bounds__(256) void gat_gemm_wmma(
        const float* __restrict__ h, const float* __restrict__ W,
        const float* __restrict__ bias, float* __restrict__ hp, int n_nodes) {
    const int lane = threadIdx.x & 31;
    const int wave = threadIdx.x >> 5;
    const int half = lane >> 4;      // 0: lanes 0-15, 1: lanes 16-31
    const int l15  = lane & 15;

    const int m0 = blockIdx.x * 16;  // node-row tile base
    const int n0 = wave * 16;        // output-column tile base

    int arow = m0 + l15;
    if (!FULL_TILE && arow >= n_nodes) arow = n_nodes - 1;  // clamp, stores predicated
    const float* aptr = h + (size_t)arow * D_IN + half * 2;

    const int col = n0 + l15;
    const float* bptr = W + (size_t)(half * 2) * D_OUT + col;

    v8f acc = {};
    #pragma unroll 8
    for (int k = 0; k < D_IN; k += 4) {
        v2f a = *(const v2f*)(aptr + k);           // b64: K = k+2*half, +1
        v2f bm;
        bm[0] = bptr[(size_t)k * D_OUT];           // row k + 2*half
        bm[1] = bptr[(size_t)k * D_OUT + D_OUT];   // row k + 2*half + 1
        acc = __builtin_amdgcn_wmma_f32_16x16x4_f32(
            /*neg_a=*/false, a, /*neg_b=*/false, bm,
            /*c_mod=*/(short)0, acc, /*reuse_a=*/false, /*reuse_b=*/false);
    }

    const float bc = bias[col];
    float* op = hp + (size_t)(m0 + half * 8) * D_OUT + col;
    #pragma unroll
    for (int i = 0; i < 8; ++i) {
        if (FULL_TILE || (m0 + i + half * 8) < n_nodes)
            op[(size_t)i * D_OUT] = acc[i] + bc;
    }
}

// ---------------------------------------------------------------------------
// Per-edge attention logit: e = leaky_relu(hp[src].wa_s + hp[dst].wa_d + b_att)
// plus atomic segment-max into emax[dst]. One wave32 per edge; each lane owns
// one contiguous float4 (global_load_b128 gathers), shfl-xor tree reduction.
// hp (51 MB) lives in the 192 MB L2.
// ---------------------------------------------------------------------------
__global__ __launch_bounds__(256) void gat_edge_att(
        const float* __restrict__ hp, const int* __restrict__ src,
        const int* __restrict__ dst, const float* __restrict__ w_att,
        const float* __restrict__ b_att, float* __restrict__ e_arr,
        float* __restrict__ emax, int n_edges) {
    const int lane = threadIdx.x & 31;
    const int wave = threadIdx.x >> 5;
    const int edge = blockIdx.x * 8 + wave;
    if (edge >= n_edges) return;

    const int s = src[edge];
    const int d = dst[edge];
    const int c0 = lane * 4;                      // 32 lanes x 4 = 128

    const float4 hs = *(const float4*)(hp + (size_t)s * D_OUT + c0);
    const float4 hd = *(const float4*)(hp + (size_t)d * D_OUT + c0);
    const float4 ws = *(const float4*)(w_att + c0);
    const float4 wd = *(const float4*)(w_att + D_OUT + c0);

    float acc;
    acc = fmaf(hs.x, ws.x, hd.x * wd.x);
    acc = fmaf(hs.y, ws.y, fmaf(hd.y, wd.y, acc));
    acc = fmaf(hs.z, ws.z, fmaf(hd.z, wd.z, acc));
    acc = fmaf(hs.w, ws.w, fmaf(hd.w, wd.w, acc));

    #pragma unroll
    for (int off = 16; off > 0; off >>= 1)
        acc += __shfl_xor(acc, off, 32);

    if (lane == 0) {
        float ev = acc + b_att[0];
        ev = (ev > 0.0f) ? ev : 0.01f * ev;       // jax leaky_relu slope 0.01
        e_arr[edge] = ev;
        atomic_max_f32(&emax[d], ev);
    }
}

// ---------------------------------------------------------------------------
// ex = exp(e - emax[dst]); denom[dst] += ex; out[dst,:] += ex * hp[src,:].
// One wave32 per edge; b128 gather of hp[src], 4 f32 atomic adds per lane
// (atomics land in L2 since out is 51 MB).
// ---------------------------------------------------------------------------
__global__ __launch_bounds__(256) void gat_edge_scatter(
        const float* __restrict__ hp, const int* __restrict__ src,
        const int* __restrict__ dst, const float* __restrict__ e_arr,
        const float* __restrict__ emax, float* __restrict__ denom,
        float* __restrict__ out, int n_edges) {
    const int lane = threadIdx.x & 31;
    const int wave = threadIdx.x >> 5;
    const int edge = blockIdx.x * 8 + wave;
    if (edge >= n_edges) return;

    const int s = src[edge];
    const int d = dst[edge];
    const float ex = __expf(e_arr[edge] - emax[d]);

    if (lane == 0) atomicAdd(&denom[d], ex);

    const int c0 = lane * 4;
    const float4 hv = *(const float4*)(hp + (size_t)s * D_OUT + c0);
    float* od = out + (size_t)d * D_OUT + c0;
    atomicAdd(od + 0, ex * hv.x);
    atomicAdd(od + 1, ex * hv.y);
    atomicAdd(od + 2, ex * hv.z);
    atomicAdd(od + 3, ex * hv.w);
}

// ---------------------------------------------------------------------------
// out = num / max(denom, guard), float4 per thread (b128 load+store).
// ---------------------------------------------------------------------------
__global__ __launch_bounds__(256) void gat_final(float4* __restrict__ out4,
                                                 const float* __restrict__ denom,
                                                 int n_nodes) {
    int i = blockIdx.x * 256 + threadIdx.x;
    if (i < n_nodes * (D_OUT / 4)) {
        float dn = denom[i >> 5];                 // (i*4) / 128
        float r  = (dn > 0.0f) ? (1.0f / dn) : 1.0f;
        float4 v = out4[i];
        v.x *= r; v.y *= r; v.z *= r; v.w *= r;
        out4[i] = v;
    }
}

// ---------------------------------------------------------------------------
extern "C" void kernel_launch(void* const* d_in, const int* in_sizes, int n_in,
                              void* d_out, int out_size, void* d_ws, size_t ws_size,
                              hipStream_t stream) {
    const float* h     = (const float*)d_in[0];
    const int*   src   = (const int*)  d_in[1];
    const int*   dst   = (const int*)  d_in[2];
    const float* W     = (const float*)d_in[3];
    const float* b     = (const float*)d_in[4];
    const float* w_att = (const float*)d_in[5];
    const float* b_att = (const float*)d_in[6];
    float* out = (float*)d_out;

    const int n_nodes = in_sizes[0] / D_IN;
    const int n_edges = in_sizes[1];

    // Workspace carve-up: hp | e | emax | denom  (~58.4 MB total)
    float* hp    = (float*)d_ws;
    float* e_arr = hp + (size_t)n_nodes * D_OUT;
    float* emax  = e_arr + n_edges;
    float* denom = emax + n_nodes;

    const int total4 = n_nodes * (D_OUT / 4);

    gat_init<<<(total4 + 255) / 256, 256, 0, stream>>>((float4*)out, emax, denom,
                                                       n_nodes);

    const int gemm_blocks = (n_nodes + 15) / 16;
    if ((n_nodes & 15) == 0) {
        gat_gemm_wmma<true><<<gemm_blocks, 256, 0, stream>>>(h, W, b, hp, n_nodes);
    } else {
        gat_gemm_wmma<false><<<gemm_blocks, 256, 0, stream>>>(h, W, b, hp, n_nodes);
    }

    gat_edge_att<<<(n_edges + 7) / 8, 256, 0, stream>>>(hp, src, dst, w_att, b_att,
                                                        e_arr, emax, n_edges);
    gat_edge_scatter<<<(n_edges + 7) / 8, 256, 0, stream>>>(hp, src, dst, e_arr,
                                                            emax, denom, out, n_edges);
    gat_final<<<(total4 + 255) / 256, 256, 0, stream>>>((float4*)out, denom, n_nodes);
}